// Scalar3DHMM_67662914781352
// MI455X (gfx1250) — compile-verified
//
#include <hip/hip_runtime.h>
#include <hip/hip_bf16.h>
#include <stdint.h>

// Problem constants (match setup_inputs): B=32, T=64, L=16, S=8*32*32=8192, V=8192.
#define S_TOT 8192
#define V_TOT 8192
#define B_N   32
#define T_N   64
#define L_N   16

typedef __attribute__((ext_vector_type(2))) float v2f;
typedef __attribute__((ext_vector_type(8))) float v8f;
typedef __attribute__((ext_vector_type(4))) int   v4i;

// ---------------- online logsumexp helpers ----------------
__device__ __forceinline__ void lse_acc(float& m, float& s, float v) {
  if (v <= m) {
    s += __expf(v - m);
  } else {
    s = s * __expf(m - v) + 1.0f;
    m = v;
  }
}
__device__ __forceinline__ void lse_join(float& m, float& s, float m2, float s2) {
  if (m2 <= m) {
    s += s2 * __expf(m2 - m);
  } else {
    s = s * __expf(m - m2) + s2;
    m = m2;
  }
}

// ---------------- CDNA5 async global->LDS copy (ASYNCcnt path) ----------------
// Builtin signature (from hipcc diagnostic): (as1 v4i* src, as3 v4i* dst, imm, imm).
__device__ __forceinline__ void async_load_b128(const void* gsrc, void* ldst) {
#if __has_builtin(__builtin_amdgcn_global_load_async_to_lds_b128)
  __builtin_amdgcn_global_load_async_to_lds_b128(
      (__attribute__((address_space(1))) v4i*)(uintptr_t)gsrc,
      (__attribute__((address_space(3))) v4i*)(uint32_t)(uintptr_t)ldst,
      0, 0);
#else
  unsigned lds = (unsigned)(uintptr_t)ldst;  // low 32 bits of flat LDS addr = LDS offset
  asm volatile("global_load_async_to_lds_b128 %0, %1, off"
               :: "v"(lds), "v"(gsrc) : "memory");
#endif
}
__device__ __forceinline__ void wait_async0() {
#if __has_builtin(__builtin_amdgcn_s_wait_asynccnt)
  __builtin_amdgcn_s_wait_asynccnt(0);
#else
  asm volatile("s_wait_asynccnt 0" ::: "memory");
#endif
}

// ================= Kernel 1: per-row logsumexp of emission (S rows x V) =================
__global__ __launch_bounds__(256) void k_row_lse(const float* __restrict__ em,
                                                 float* __restrict__ lse_row) {
  __shared__ float rm[256], rs[256];
  const int row = blockIdx.x;
  const float* r = em + (size_t)row * V_TOT;
  float m = -3.0e38f, s = 0.0f;
  for (int i = threadIdx.x; i < V_TOT; i += 256) lse_acc(m, s, r[i]);
  rm[threadIdx.x] = m; rs[threadIdx.x] = s;
  __syncthreads();
  for (int st = 128; st > 0; st >>= 1) {
    if (threadIdx.x < st) {
      float mm = rm[threadIdx.x], ss = rs[threadIdx.x];
      lse_join(mm, ss, rm[threadIdx.x + st], rs[threadIdx.x + st]);
      rm[threadIdx.x] = mm; rs[threadIdx.x] = ss;
    }
    __syncthreads();
  }
  if (threadIdx.x == 0) lse_row[row] = rm[0] + __logf(rs[0]);
}

// ====== Kernel 2: em_T[v][s] = em[s][v] - lse_row[s]  (LDS tiled transpose) ======
__global__ __launch_bounds__(256) void k_transpose(const float* __restrict__ em,
                                                   const float* __restrict__ lse_row,
                                                   float* __restrict__ emT) {
  __shared__ float tile[32][33];
  const int v0 = blockIdx.x * 32;
  const int s0 = blockIdx.y * 32;
  const int tx = threadIdx.x, ty = threadIdx.y;   // block (32, 8)
  for (int j = 0; j < 32; j += 8)
    tile[ty + j][tx] = em[(size_t)(s0 + ty + j) * V_TOT + v0 + tx];
  __syncthreads();
  const float l = lse_row[s0 + tx];
  for (int j = 0; j < 32; j += 8)
    emT[(size_t)(v0 + ty + j) * S_TOT + s0 + tx] = tile[tx][ty + j] - l;
}

// ================= Kernel 3: row log_softmax of transitions (S x 7) =================
__global__ __launch_bounds__(256) void k_log_tr(const float* __restrict__ tr,
                                                float* __restrict__ ltr) {
  const int s = blockIdx.x * 256 + threadIdx.x;
  if (s >= S_TOT) return;
  float v[7]; float m = -3.0e38f;
  for (int k = 0; k < 7; ++k) { v[k] = tr[s * 7 + k]; m = fmaxf(m, v[k]); }
  float sum = 0.0f;
  for (int k = 0; k < 7; ++k) sum += __expf(v[k] - m);
  const float l = m + __logf(sum);
  for (int k = 0; k < 7; ++k) ltr[s * 7 + k] = v[k] - l;
}

// ===== Kernel 4: emit table E[(t*B+b)][s] = sum_l em_T[tok(b,t,l)][s]  (WMMA) =====
// 16 (t,b)-pairs per block (x), 8 waves x 16-state tiles per block (y).
// E = A x G: A[m,j] = (j>>4 == m) constant 0/1 selector, G[j,s] = em_T[tok_j][s].
__global__ __launch_bounds__(256) void k_emit(const int* __restrict__ stories,
                                              const float* __restrict__ emT,
                                              float* __restrict__ E) {
  __shared__ int tok[256];  // [pair(0..15)][l(0..15)]
  const int tid = threadIdx.x;
  {
    const int p = tid >> 4, l = tid & 15;
    const int P = blockIdx.x * 16 + p;       // pair id = t*B + b
    const int t = P >> 5, b = P & 31;
    tok[tid] = stories[((size_t)b * T_N + t) * L_N + l];
  }
  __syncthreads();
  const int lane  = tid & 31;                 // wave32
  const int wave  = tid >> 5;
  const int s0    = (blockIdx.y * 8 + wave) * 16;
  const int n     = lane & 15;
  const int khalf = (lane >> 4) * 2;
#if __has_builtin(__builtin_amdgcn_wmma_f32_16x16x4_f32)
  const int m = lane & 15;                    // A-matrix row for this lane
  v8f c = {};
  for (int kk = 0; kk < 256; kk += 4) {
    const int j0 = kk + khalf, j1 = j0 + 1;
    v2f a, bb;
    a.x = ((j0 >> 4) == m) ? 1.0f : 0.0f;
    a.y = ((j1 >> 4) == m) ? 1.0f : 0.0f;
    bb.x = emT[(size_t)tok[j0] * S_TOT + s0 + n];
    bb.y = emT[(size_t)tok[j1] * S_TOT + s0 + n];
    c = __builtin_amdgcn_wmma_f32_16x16x4_f32(false, a, false, bb,
                                              (short)0, c, false, false);
  }
  for (int r = 0; r < 8; ++r) {
    const int mr = (lane < 16) ? r : (r + 8);   // D layout: vgpr r -> rows r / r+8
    const int P  = blockIdx.x * 16 + mr;
    E[(size_t)P * S_TOT + s0 + n] = c[r];
  }
#else
  const int half = lane >> 4;
  for (int p = half * 8; p < half * 8 + 8; ++p) {
    float acc = 0.0f;
    for (int l = 0; l < 16; ++l)
      acc += emT[(size_t)tok[p * 16 + l] * S_TOT + s0 + n];
    const int P = blockIdx.x * 16 + p;
    E[(size_t)P * S_TOT + s0 + n] = acc;
  }
#endif
}

// ================= Kernel 5: forward recurrence, one workgroup per batch =================
// LDS: alpha double buffer (2x32KB) + emit-row double buffer (2x32KB, filled with
// async global->LDS b128 copies) + reduction arrays (8KB) = 136KB per WG.
__device__ __forceinline__ float block_lse(float m, float s, float* rm, float* rs, int tid) {
  __syncthreads();                 // protect rm/rs reuse across calls
  rm[tid] = m; rs[tid] = s;
  __syncthreads();
  for (int st = 512; st > 0; st >>= 1) {
    if (tid < st) {
      float mm = rm[tid], ss = rs[tid];
      lse_join(mm, ss, rm[tid + st], rs[tid + st]);
      rm[tid] = mm; rs[tid] = ss;
    }
    __syncthreads();
  }
  return rm[0] + __logf(rs[0]);
}

__global__ __launch_bounds__(1024) void k_forward(const int* __restrict__ lengths,
                                                  const float* __restrict__ prior,
                                                  const float* __restrict__ ltr,
                                                  const float* __restrict__ E,
                                                  float* __restrict__ out) {
  extern __shared__ float smem[];
  float* alpha0 = smem;                 // 8192
  float* alpha1 = smem + 8192;          // 8192
  float* eb0    = smem + 16384;         // 8192 (emit row buffer 0)
  float* eb1    = smem + 24576;         // 8192 (emit row buffer 1)
  float* rm     = smem + 32768;         // 1024
  float* rs     = smem + 33792;         // 1024

  const int b   = blockIdx.x;
  const int tid = threadIdx.x;
  const int idxb = min(max(lengths[b] - 1, 0), T_N - 1);

  // log_softmax of prior: lse over 8192 values
  float pm = -3.0e38f, ps = 0.0f;
  for (int i = 0; i < 8; ++i) lse_acc(pm, ps, prior[tid + i * 1024]);
  const float lse_prior = block_lse(pm, ps, rm, rs, tid);

  // alpha_0 = E[t=0,b,:] + log_prior
  const float* E0 = E + (size_t)b * S_TOT;   // pair id 0*B + b
  float tm = -3.0e38f, ts = 0.0f;
  for (int i = 0; i < 8; ++i) {
    const int st = tid + i * 1024;
    const float a0 = E0[st] + prior[st] - lse_prior;
    alpha0[st] = a0;
    lse_acc(tm, ts, a0);
  }
  // prefetch emit row t=1 into eb1 (2 x b128 per lane = 32KB total)
  {
    const float* E1 = E + (size_t)(1 * B_N + b) * S_TOT;
    async_load_b128(E1 + tid * 4,        eb1 + tid * 4);
    async_load_b128(E1 + 4096 + tid * 4, eb1 + 4096 + tid * 4);
  }
  const float tot0 = block_lse(tm, ts, rm, rs, tid);
  if (tid == 0 && idxb == 0) out[b] = tot0;

  for (int t = 1; t < T_N; ++t) {
    const int cur = t & 1;
    float* an        = cur ? alpha1 : alpha0;   // write
    const float* ap  = cur ? alpha0 : alpha1;   // read (previous step)
    const float* ecur = cur ? eb1 : eb0;        // emit row for this step
    float* enext      = cur ? eb0 : eb1;        // buffer for next step's row
    wait_async0();          // this wave's async loads of ecur done
    __syncthreads();        // all waves' loads done; prev-step eb consumption done
    if (t + 1 < T_N) {      // prefetch next emit row into the other buffer
      const float* En = E + (size_t)((t + 1) * B_N + b) * S_TOT;
      async_load_b128(En + tid * 4,        enext + tid * 4);
      async_load_b128(En + 4096 + tid * 4, enext + 4096 + tid * 4);
    }
    float qm = -3.0e38f, qs = 0.0f;
    for (int i = 0; i < 8; ++i) {
      const int st = tid + i * 1024;
      const int x = st & 31, y = (st >> 5) & 31, z = st >> 10;
      // 7-neighbor log-semiring gather: contributions from source states
      float v[7]; int nv = 0;
      v[nv++] = ap[st] + ltr[st * 7 + 0];                              // stay
      if (x > 0)  { const int q = st - 1;    v[nv++] = ap[q] + ltr[q * 7 + 1]; }
      if (x < 31) { const int q = st + 1;    v[nv++] = ap[q] + ltr[q * 7 + 2]; }
      if (y > 0)  { const int q = st - 32;   v[nv++] = ap[q] + ltr[q * 7 + 3]; }
      if (y < 31) { const int q = st + 32;   v[nv++] = ap[q] + ltr[q * 7 + 4]; }
      if (z > 0)  { const int q = st - 1024; v[nv++] = ap[q] + ltr[q * 7 + 5]; }
      if (z > 1)  { const int q = st - 2048; v[nv++] = ap[q] + ltr[q * 7 + 6]; }
      float mx = v[0];
      for (int j = 1; j < nv; ++j) mx = fmaxf(mx, v[j]);
      float sm = 0.0f;
      for (int j = 0; j < nv; ++j) sm += __expf(v[j] - mx);
      const float nx = mx + __logf(sm) + ecur[st];
      an[st] = nx;
      lse_acc(qm, qs, nx);
    }
    const float tt = block_lse(qm, qs, rm, rs, tid);
    if (tid == 0 && t == idxb) out[b] = tt;
  }
}

// =============================== host entry ===============================
extern "C" void kernel_launch(void* const* d_in, const int* in_sizes, int n_in,
                              void* d_out, int out_size, void* d_ws, size_t ws_size,
                              hipStream_t stream) {
  (void)in_sizes; (void)n_in; (void)out_size; (void)ws_size;
  const int*   stories = (const int*)d_in[0];
  const int*   lengths = (const int*)d_in[1];
  const float* em      = (const float*)d_in[2];
  const float* tr      = (const float*)d_in[3];
  const float* prior   = (const float*)d_in[4];
  // d_in[5]=xy_size(32), d_in[6]=z_size(8): fixed by the reference; baked into constants.

  char* ws = (char*)d_ws;
  float* emT = (float*)(ws);                                   // 256 MB transposed log-emission
  float* E   = (float*)(ws + (size_t)268435456);               //  64 MB emit table (T*B, S)
  float* lse = (float*)(ws + (size_t)268435456 + 67108864);    //  32 KB per-row lse
  float* ltr = (float*)(ws + (size_t)268435456 + 67108864 + 32768);  // 224 KB log transitions
  float* out = (float*)d_out;

  k_log_tr  <<<32,   256, 0, stream>>>(tr, ltr);
  k_row_lse <<<8192, 256, 0, stream>>>(em, lse);
  k_transpose<<<dim3(256, 256), dim3(32, 8), 0, stream>>>(em, lse, emT);
  k_emit    <<<dim3(128, 64), 256, 0, stream>>>(stories, emT, E);
  k_forward <<<32, 1024, 34816 * sizeof(float), stream>>>(lengths, prior, ltr, E, out);
}